// ResidualE8LatticeVQ_43224550867403
// MI455X (gfx1250) — compile-verified
//
#include <hip/hip_runtime.h>

#ifndef __has_builtin
#define __has_builtin(x) 0
#endif

#define AS1 __attribute__((address_space(1)))
#define AS3 __attribute__((address_space(3)))

static constexpr int kLevels = 6;

// int32x4 vector type matching the async-LDS builtin's pointee type.
typedef int v4i __attribute__((vector_size(4 * sizeof(int))));

// 16-byte vector with only 4-byte alignment guarantee (indices region is
// offset by the scalar loss slot, so it is 4-byte aligned only).
typedef float f4a4 __attribute__((ext_vector_type(4), aligned(4)));

// ---------------------------------------------------------------------------
// gfx1250 async global->LDS copy (ASYNCcnt path) + wait
// ---------------------------------------------------------------------------
__device__ __forceinline__ void async_copy_b128(const float* gsrc, float* ldst) {
#if __has_builtin(__builtin_amdgcn_global_load_async_to_lds_b128)
  // Signature (probe-confirmed): (AS1 v4i* gsrc, AS3 v4i* lds, Imm off, Imm cpol)
  __builtin_amdgcn_global_load_async_to_lds_b128(
      (AS1 v4i*)gsrc, (AS3 v4i*)ldst, 0, 0);
#else
  // GV mode: vdst = 32-bit LDS byte offset, vaddr = 64-bit global address.
  unsigned lds_off = (unsigned)(unsigned long long)(__SIZE_TYPE__)ldst;
  unsigned long long ga = (unsigned long long)gsrc;
  asm volatile("global_load_async_to_lds_b128 %0, %1, off"
               :: "v"(lds_off), "v"(ga) : "memory");
#endif
}

__device__ __forceinline__ void wait_async0() {
#if __has_builtin(__builtin_amdgcn_s_wait_asynccnt)
  __builtin_amdgcn_s_wait_asynccnt(0);
#else
  asm volatile("s_wait_asynccnt 0x0" ::: "memory");
#endif
}

// ---------------------------------------------------------------------------
// E8 lattice math (matches jnp reference: RNE rounding, first-index argmax)
// ---------------------------------------------------------------------------
__device__ __forceinline__ void nearest_d8(const float v[8], float f[8]) {
  float delta[8];
#pragma unroll
  for (int i = 0; i < 8; ++i) {
    f[i] = rintf(v[i]);            // v_rndne_f32: round half to even
    delta[i] = v[i] - f[i];
  }
  int k = 0;
  float best = fabsf(delta[0]);
#pragma unroll
  for (int i = 1; i < 8; ++i) {    // strict > keeps first max (jnp.argmax)
    float a = fabsf(delta[i]);
    if (a > best) { best = a; k = i; }
  }
  float fs = 0.0f;
#pragma unroll
  for (int i = 0; i < 8; ++i) fs += f[i];
  const bool odd = (((int)fs) & 1) != 0;
#pragma unroll
  for (int i = 0; i < 8; ++i) {    // select-based correction, no dyn indexing
    float st = (delta[i] >= 0.0f) ? 1.0f : -1.0f;
    f[i] += (odd && (i == k)) ? st : 0.0f;
  }
}

__device__ __forceinline__ void nearest_e8(const float v[8], float y[8]) {
  float y0[8], y1[8], vm[8];
  nearest_d8(v, y0);
#pragma unroll
  for (int i = 0; i < 8; ++i) vm[i] = v[i] - 0.5f;
  nearest_d8(vm, y1);
#pragma unroll
  for (int i = 0; i < 8; ++i) y1[i] += 0.5f;
  float d0 = 0.0f, d1 = 0.0f;
#pragma unroll
  for (int i = 0; i < 8; ++i) {
    float e0 = v[i] - y0[i]; d0 += e0 * e0;
    float e1 = v[i] - y1[i]; d1 += e1 * e1;
  }
  const bool pick0 = (d0 <= d1);
#pragma unroll
  for (int i = 0; i < 8; ++i) y[i] = pick0 ? y0[i] : y1[i];
}

// ---------------------------------------------------------------------------
// Main kernel: one thread per 8-vector; 256 threads (8 wave32) per block.
// LDS: 8KB input tile (async-staged) + 8KB hash bitmap + 1KB reduction.
// ---------------------------------------------------------------------------
__global__ __launch_bounds__(256) void e8rvq_main_kernel(
    const float* __restrict__ x,
    float* __restrict__ quant,
    float* __restrict__ indices,
    float* __restrict__ loss_partial,
    unsigned int* __restrict__ gbitmap,
    int n_vec) {
  __shared__ __align__(16) float tile[256 * 8];
  __shared__ unsigned int sbm[2048];   // 65536-bit block-local hash bitmap
  __shared__ float red[256];

  const int tid = threadIdx.x;
  const long long gvec = (long long)blockIdx.x * 256 + tid;
  const bool active = gvec < n_vec;

  // Stage this block's 8KB input tile through the CDNA5 async LDS path.
  if (active) {
    const float* gsrc = x + gvec * 8;
    async_copy_b128(gsrc, &tile[tid * 8]);
    async_copy_b128(gsrc + 4, &tile[tid * 8 + 4]);
  }
  // Zero the block-local bitmap while the async copies are in flight.
#pragma unroll
  for (int j = 0; j < 8; ++j) sbm[tid + j * 256] = 0u;
  wait_async0();
  __syncthreads();

  // scale[l] = max(2 * 240^(-l/2), 1e-6)  (DECAY = sqrt(240); none clamp)
  constexpr float SC[kLevels] = {
      2.0f, 0.12909944487358058f, 0.008333333333333333f,
      5.3791435363991164e-4f, 3.4722222222222224e-5f, 2.2413098129371e-6f};
  constexpr float IS[kLevels] = {
      0.5f, 7.745966692414834f, 120.0f,
      1859.0320061795602f, 28800.0f, 446167.6814830944f};

  float loss = 0.0f;
  if (active) {
    float xv[8], qs[8], r[8];
    const float4 a = *(const float4*)&tile[tid * 8];
    const float4 b = *(const float4*)&tile[tid * 8 + 4];
    xv[0] = a.x; xv[1] = a.y; xv[2] = a.z; xv[3] = a.w;
    xv[4] = b.x; xv[5] = b.y; xv[6] = b.z; xv[7] = b.w;
#pragma unroll
    for (int i = 0; i < 8; ++i) { qs[i] = 0.0f; r[i] = xv[i]; }

    float* idx_base = indices + gvec * (kLevels * 8);
#pragma unroll
    for (int l = 0; l < kLevels; ++l) {
      const float s = SC[l], inv = IS[l];
      float xs[8], y[8];
#pragma unroll
      for (int i = 0; i < 8; ++i) xs[i] = r[i] * inv;
      nearest_e8(xs, y);
      int hash = 0;
      float cf[8];
#pragma unroll
      for (int i = 0; i < 8; ++i) {
        int c = (int)rintf(2.0f * y[i]);   // codes = round(2*y), exact
        hash += c;
        cf[i] = (float)c;
        qs[i] += y[i] * s;                 // qsum += y_hard
      }
#pragma unroll
      for (int i = 0; i < 8; ++i) r[i] = xv[i] - qs[i];  // residual = x - qsum
      f4a4 c0 = {cf[0], cf[1], cf[2], cf[3]};
      f4a4 c1 = {cf[4], cf[5], cf[6], cf[7]};
      *(f4a4*)(idx_base + l * 8) = c0;
      *(f4a4*)(idx_base + l * 8 + 4) = c1;
      int h = hash + 32768;
      h = h < 0 ? 0 : (h > 65535 ? 65535 : h);
      atomicOr(&sbm[h >> 5], 1u << (h & 31));            // ds_or_b32
    }
#pragma unroll
    for (int i = 0; i < 8; ++i) loss += r[i] * r[i];
    float4 q0 = make_float4(qs[0], qs[1], qs[2], qs[3]);
    float4 q1 = make_float4(qs[4], qs[5], qs[6], qs[7]);
    *(float4*)(quant + gvec * 8) = q0;
    *(float4*)(quant + gvec * 8 + 4) = q1;
  }

  // Deterministic per-block loss reduction (fixed LDS tree).
  red[tid] = loss;
  __syncthreads();
#pragma unroll
  for (int st = 128; st > 0; st >>= 1) {
    if (tid < st) red[tid] += red[tid + st];
    __syncthreads();
  }
  if (tid == 0) loss_partial[blockIdx.x] = red[0];

  // Merge block bitmap into global bitmap (order-independent => deterministic).
#pragma unroll
  for (int j = 0; j < 8; ++j) {
    unsigned w = sbm[tid + j * 256];
    if (w) atomicOr(&gbitmap[tid + j * 256], w);
  }
}

// ---------------------------------------------------------------------------
// Finalize: fixed-order reduce of loss partials + popcount of hash bitmap.
// ---------------------------------------------------------------------------
__global__ __launch_bounds__(256) void e8rvq_finalize_kernel(
    const float* __restrict__ loss_partial, int n_blocks,
    const unsigned int* __restrict__ gbitmap,
    float* __restrict__ loss_out, float* __restrict__ ppx_out,
    float inv_elems) {
  __shared__ float redf[256];
  __shared__ int redi[256];
  const int tid = threadIdx.x;
  float s = 0.0f;
  for (int i = tid; i < n_blocks; i += 256) s += loss_partial[i];
  int c = 0;
  for (int i = tid; i < 2048; i += 256) c += __popc(gbitmap[i]);
  redf[tid] = s;
  redi[tid] = c;
  __syncthreads();
#pragma unroll
  for (int st = 128; st > 0; st >>= 1) {
    if (tid < st) { redf[tid] += redf[tid + st]; redi[tid] += redi[tid + st]; }
    __syncthreads();
  }
  if (tid == 0) {
    loss_out[0] = redf[0] * inv_elems;   // mean((x - quantized)^2)
    ppx_out[0] = (float)redi[0];         // # distinct hash values
  }
}

// ---------------------------------------------------------------------------
// Launcher. d_out layout (all f32, reference return order):
//   [quant: N][loss: 1][indices: n_vec*6*8][perplexity: 1]
// d_ws layout: [8KB hash bitmap][n_blocks f32 loss partials]
// ---------------------------------------------------------------------------
extern "C" void kernel_launch(void* const* d_in, const int* in_sizes, int n_in,
                              void* d_out, int out_size, void* d_ws,
                              size_t ws_size, hipStream_t stream) {
  (void)n_in; (void)out_size; (void)ws_size;
  const float* x = (const float*)d_in[0];
  const long long n_elems = in_sizes[0];            // 16,777,216
  const int n_vec = (int)(n_elems / 8);             // 2,097,152

  float* out = (float*)d_out;
  float* quant = out;
  float* loss = out + n_elems;
  float* indices = out + n_elems + 1;
  float* ppx = out + n_elems + 1 + (long long)n_vec * kLevels * 8;

  unsigned int* gbitmap = (unsigned int*)d_ws;
  float* partials = (float*)((char*)d_ws + 2048 * sizeof(unsigned int));

  (void)hipMemsetAsync(d_ws, 0, 2048 * sizeof(unsigned int), stream);

  const int blocks = (n_vec + 255) / 256;           // 8192
  e8rvq_main_kernel<<<blocks, 256, 0, stream>>>(x, quant, indices, partials,
                                                gbitmap, n_vec);
  e8rvq_finalize_kernel<<<1, 256, 0, stream>>>(partials, blocks, gbitmap, loss,
                                               ppx, 1.0f / (float)n_elems);
}